// Centroids_21380347199545
// MI455X (gfx1250) — compile-verified
//
#include <hip/hip_runtime.h>
#include <cstdint>

// ---------------------------------------------------------------------------
// Segment-mean + EMA centroids for MI455X (gfx1250).
// Bandwidth-bound: x (1.02 GB) crosses HBM exactly once; random-class scatter
// handled by LDS ds_add_f32; x rows staged via the CDNA5 async-to-LDS path
// with double-buffered staging and partial ASYNCcnt waits to pipeline the
// copy latency; small global f32-atomic flush per block.
// ---------------------------------------------------------------------------

#define FEATURE_DIM    128
#define NUM_CLASSES    1000
#define DECAY_C        0.3f

#define CPG            96            // classes per group (LDS table 48KB)
#define PASSES         11            // ceil(1000/96)
#define ROWS_PER_CHUNK 16384
#define BLOCK          128           // 4 wave32 waves
#define NW             (BLOCK / 32)
#define SLOTS          3             // rows per async batch
#define NBUF           2             // double-buffered staging

__device__ __forceinline__ uint32_t lds_offset(const void* shared_ptr) {
  // Flat shared-aperture pointer: addr[31:0] is the LDS byte offset.
  return (uint32_t)(uintptr_t)shared_ptr;
}

__device__ __forceinline__ void async_copy_row16(uint32_t lds_off, const float* gsrc) {
  // Each lane copies 16 bytes from global address gsrc to LDS offset lds_off, tracked on ASYNCcnt.
  asm volatile("global_load_async_to_lds_b128 %0, %1, off"
               :
               : "v"(lds_off), "v"((uint64_t)(uintptr_t)gsrc)
               : "memory");
}

__device__ __forceinline__ void wait_async_le(int n) {
  // Async loads retire in order (ISA 08 §4.1): waiting ASYNCcnt<=n with n ops
  // just issued guarantees all older async copies have landed in LDS.
  switch (n) {
    case 0:  asm volatile("s_wait_asynccnt 0" ::: "memory"); break;
    case 1:  asm volatile("s_wait_asynccnt 1" ::: "memory"); break;
    case 2:  asm volatile("s_wait_asynccnt 2" ::: "memory"); break;
    default: asm volatile("s_wait_asynccnt 3" ::: "memory"); break;
  }
}

__global__ __launch_bounds__(BLOCK) void centroid_accum(
    const float* __restrict__ x, const int* __restrict__ y,
    const unsigned char* __restrict__ ymask,
    float* __restrict__ gsums, int* __restrict__ gcntM, int* __restrict__ gcntA,
    int n)
{
  __shared__ float tbl[CPG * FEATURE_DIM];                    // 49152 B
  __shared__ int   cntM[CPG];                                 //   384 B
  __shared__ int   cntA[CPG];                                 //   384 B
  __shared__ float stage[NW * NBUF * SLOTS * FEATURE_DIM];    // 12288 B (62208 total)

  const int g     = blockIdx.x % PASSES;     // class group minor -> 11 siblings
  const int chunk = blockIdx.x / PASSES;     // share y/mask chunk through L2
  const int cs    = chunk * ROWS_PER_CHUNK;
  const int ceLim = min(n, cs + ROWS_PER_CHUNK);
  const int wave  = threadIdx.x >> 5;
  const int lane  = threadIdx.x & 31;
  const int cbase = g * CPG;

  for (int i = threadIdx.x; i < CPG * FEATURE_DIM; i += BLOCK) tbl[i] = 0.0f;
  for (int i = threadIdx.x; i < CPG; i += BLOCK) { cntM[i] = 0; cntA[i] = 0; }
  __syncthreads();

  const uint32_t stage0 = lds_offset(&stage[0]);

  // One pending async batch per wave, double-buffered staging slots.
  int pend_nb  = 0;
  int pend_buf = 0;
  int pend_cls[SLOTS];

  for (int base = cs + wave * 32; base < ceLim; base += NW * 32) {
    const int  r     = base + lane;
    const bool valid = r < ceLim;
    const int  yr    = valid ? y[r] : -1;
    const int  mk    = valid ? (ymask[r] ? 1 : 0) : 0;
    const int  cls   = yr - cbase;
    const bool inG   = valid && (cls >= 0) && (cls < CPG) && (yr < NUM_CLASSES);

    if (inG) {
      atomicAdd(&cntA[cls], 1);              // presence (mask-independent)
      if (mk) atomicAdd(&cntM[cls], 1);
    }

    unsigned mm = (unsigned)__ballot(inG && mk);
    while (mm) {
      // Collect up to SLOTS matched rows (wave-uniform).
      int nb = 0;
      int rows[SLOTS];
      int clsb[SLOTS];
      while (mm && nb < SLOTS) {
        const int src = __ffs(mm) - 1;
        mm &= mm - 1;
        rows[nb] = __shfl(r, src);
        clsb[nb] = __shfl(cls, src);
        ++nb;
      }

      // Issue new batch into the buffer NOT used by the pending batch.
      const int nbuf = pend_nb ? (pend_buf ^ 1) : 0;
#pragma unroll
      for (int s = 0; s < SLOTS; ++s) {
        if (s < nb) {
          const float*   gsrc = x + (size_t)rows[s] * FEATURE_DIM + lane * 4;
          const uint32_t off  = stage0 +
              (uint32_t)(((wave * NBUF + nbuf) * SLOTS + s) * FEATURE_DIM + lane * 4) * 4u;
          async_copy_row16(off, gsrc);
        }
      }

      // Consume the pending batch while the new one is still in flight.
      if (pend_nb) {
        wait_async_le(nb);
#pragma unroll
        for (int s = 0; s < SLOTS; ++s) {
          if (s < pend_nb) {
            const float4 v = *reinterpret_cast<const float4*>(
                &stage[((wave * NBUF + pend_buf) * SLOTS + s) * FEATURE_DIM + lane * 4]);
            float* t = &tbl[pend_cls[s] * FEATURE_DIM + lane * 4];
            atomicAdd(t + 0, v.x);
            atomicAdd(t + 1, v.y);
            atomicAdd(t + 2, v.z);
            atomicAdd(t + 3, v.w);
          }
        }
      }

      pend_nb  = nb;
      pend_buf = nbuf;
#pragma unroll
      for (int s = 0; s < SLOTS; ++s) pend_cls[s] = clsb[s];
    }
  }

  // Drain the last pending batch.
  if (pend_nb) {
    wait_async_le(0);
#pragma unroll
    for (int s = 0; s < SLOTS; ++s) {
      if (s < pend_nb) {
        const float4 v = *reinterpret_cast<const float4*>(
            &stage[((wave * NBUF + pend_buf) * SLOTS + s) * FEATURE_DIM + lane * 4]);
        float* t = &tbl[pend_cls[s] * FEATURE_DIM + lane * 4];
        atomicAdd(t + 0, v.x);
        atomicAdd(t + 1, v.y);
        atomicAdd(t + 2, v.z);
        atomicAdd(t + 3, v.w);
      }
    }
  }

  __syncthreads();

  // Flush LDS table to global accumulators (HW f32 atomics).
  for (int i = threadIdx.x; i < CPG * FEATURE_DIM; i += BLOCK) {
    const int c = cbase + (i >> 7);
    if (c < NUM_CLASSES) {
      const float v = tbl[i];
      if (v != 0.0f) unsafeAtomicAdd(&gsums[c * FEATURE_DIM + (i & 127)], v);
    }
  }
  for (int i = threadIdx.x; i < CPG; i += BLOCK) {
    const int c = cbase + i;
    if (c < NUM_CLASSES) {
      if (cntM[i]) atomicAdd(&gcntM[c], cntM[i]);
      if (cntA[i]) atomicAdd(&gcntA[c], cntA[i]);
    }
  }
}

__global__ __launch_bounds__(256) void centroid_final(
    const float* __restrict__ gsums, const int* __restrict__ gcntM,
    const int* __restrict__ gcntA, const float* __restrict__ cen,
    float* __restrict__ out, int tot)
{
  const int i = blockIdx.x * blockDim.x + threadIdx.x;
  if (i >= tot) return;
  const int   c   = i >> 7;
  const float s   = gsums[i];
  const float cm  = (float)gcntM[c];
  const float ce  = cen[i];
  const float avg = s / fmaxf(cm, 1.0f);
  out[i] = (gcntA[c] > 0) ? (DECAY_C * avg + (1.0f - DECAY_C) * ce) : ce;
}

extern "C" void kernel_launch(void* const* d_in, const int* in_sizes, int n_in,
                              void* d_out, int out_size, void* d_ws, size_t ws_size,
                              hipStream_t stream) {
  const float*         x   = (const float*)d_in[0];
  const int*           y   = (const int*)d_in[1];
  const unsigned char* ym  = (const unsigned char*)d_in[2];
  const float*         cen = (const float*)d_in[3];
  const int n = in_sizes[1];

  float* gsums = (float*)d_ws;
  int*   gcntM = (int*)((char*)d_ws + (size_t)NUM_CLASSES * FEATURE_DIM * sizeof(float));
  int*   gcntA = gcntM + NUM_CLASSES;

  const size_t tbl_bytes =
      (size_t)NUM_CLASSES * FEATURE_DIM * sizeof(float) + 2u * NUM_CLASSES * sizeof(int);
  hipMemsetAsync(d_ws, 0, tbl_bytes, stream);

  const int nchunk = (n + ROWS_PER_CHUNK - 1) / ROWS_PER_CHUNK;
  centroid_accum<<<nchunk * PASSES, BLOCK, 0, stream>>>(x, y, ym, gsums, gcntM, gcntA, n);

  const int tot = NUM_CLASSES * FEATURE_DIM;
  centroid_final<<<(tot + 255) / 256, 256, 0, stream>>>(gsums, gcntM, gcntA, cen,
                                                        (float*)d_out, tot);
}